// AnomalyAttention_86294482911883
// MI455X (gfx1250) — compile-verified
//
#include <hip/hip_runtime.h>
#include <math.h>

// AnomalyAttention for MI455X (gfx1250), wave32, WMMA fp32 path.
// B=4, L=1024, H=8, E=64. Outputs (flat, in order): V, series, prior, sigma4.

typedef __attribute__((ext_vector_type(2))) float v2f;
typedef __attribute__((ext_vector_type(4))) float v4f;
typedef __attribute__((ext_vector_type(8))) float v8f;

#define B_ 4
#define L_ 1024
#define H_ 8
#define E_ 64
#define SCALE_ 0.125f            // 1/sqrt(64)
#define NEGINF_ (-1.0e9f)
#define LDSTRIDE 1028            // padded row stride (dwords): conflict-free, 8B aligned

// ---------------------------------------------------------------------------
// Kernel 1: fused scores -> causal mask -> softmax -> series store -> V=P*values
// grid = B*H*(L/16) blocks, 128 threads (4 waves). One block = one (b,h) 16-row strip.
// ---------------------------------------------------------------------------
__global__ __launch_bounds__(128)
void attn_fused_kernel(const float* __restrict__ Q,
                       const float* __restrict__ K,
                       const float* __restrict__ V,
                       float* __restrict__ outV,
                       float* __restrict__ outSeries) {
  extern __shared__ float lds[];
  float* sm     = lds;                       // [16][LDSTRIDE] score strip
  float* red    = lds + 16 * LDSTRIDE;       // [16][8] reduction scratch
  float* rowmax = red + 128;                 // [16]
  float* rowinv = rowmax + 16;               // [16]

  const int tid  = threadIdx.x;
  const int wave = tid >> 5;
  const int lane = tid & 31;

  const int qblk = blockIdx.x & 63;          // L/16 = 64 strips
  const int bh   = blockIdx.x >> 6;          // b*H + h
  const int b    = bh >> 3;                  // H = 8
  const int h    = bh & 7;

  const int q0 = qblk * 16;
  const int m  = lane & 15;                  // M (A-side) / N (B,D-side) index
  const int hi = lane >> 4;                  // 0 -> K={0,1}, 1 -> K={2,3}

  // ---- preload Q tile as 16 A-fragments of V_WMMA_F32_16X16X4_F32 ----
  // A(16x4) layout: lanes 0-15 hold M=lane with K0/K1 in vgpr0/1,
  //                 lanes 16-31 hold M=lane-16 with K2/K3.
  v2f qa[16];
  {
    const float* qrow = Q + (((size_t)b * L_ + (q0 + m)) * H_ + h) * E_;
    #pragma unroll
    for (int kk = 0; kk < 16; ++kk)
      qa[kk] = *(const v2f*)(qrow + kk * 4 + hi * 2);
  }

  // ---- pass 1: 16x16 score tiles -> LDS (scaled + causal masked) ----
  const int iMax = q0 + 15;
  for (int t = 0; t < 16; ++t) {
    const int s0 = (wave * 16 + t) * 16;     // wave-uniform -> EXEC all-ones at WMMA
    if (s0 <= iMax) {
      const float* krow = K + (((size_t)b * L_ + (s0 + m)) * H_ + h) * E_;
      __builtin_prefetch(krow + E_ * 16, 0, 1);   // global_prefetch_b8 of key row s0+m+2
      v8f acc = {};
      #pragma unroll
      for (int kk = 0; kk < 16; ++kk) {
        v2f kb = *(const v2f*)(krow + kk * 4 + hi * 2);   // B = K^T fragment
        acc = __builtin_amdgcn_wmma_f32_16x16x4_f32(false, qa[kk], false, kb,
                                                    (short)0, acc, false, false);
      }
      // D layout: vgpr r -> row (r + 8*hi), col = s0 + m
      #pragma unroll
      for (int r = 0; r < 8; ++r) {
        const int row = r + hi * 8;
        const int i   = q0 + row;
        const int j   = s0 + m;
        const float v = (j > i) ? (SCALE_ * NEGINF_) : (SCALE_ * acc[r]);
        sm[row * LDSTRIDE + j] = v;
      }
    } else {
      #pragma unroll
      for (int r = 0; r < 8; ++r)
        sm[(r + hi * 8) * LDSTRIDE + (s0 + m)] = SCALE_ * NEGINF_;
    }
  }
  __syncthreads();

  // ---- softmax over each of the 16 rows (8 threads per row, 128 cols each) ----
  const int rrow = tid >> 3;
  const int rsub = tid & 7;
  float* rowp = &sm[rrow * LDSTRIDE + rsub * 128];

  float mx = -3.4e38f;
  for (int c = 0; c < 128; ++c) mx = fmaxf(mx, rowp[c]);
  red[rrow * 8 + rsub] = mx;
  __syncthreads();
  if (rsub == 0) {
    float m2 = red[rrow * 8];
    #pragma unroll
    for (int c = 1; c < 8; ++c) m2 = fmaxf(m2, red[rrow * 8 + c]);
    rowmax[rrow] = m2;
  }
  __syncthreads();

  const float rmax = rowmax[rrow];
  float sum = 0.f;
  for (int c = 0; c < 128; ++c) {
    const float e = __expf(rowp[c] - rmax);
    rowp[c] = e;
    sum += e;
  }
  red[rrow * 8 + rsub] = sum;
  __syncthreads();
  if (rsub == 0) {
    float s2 = 0.f;
    #pragma unroll
    for (int c = 0; c < 8; ++c) s2 += red[rrow * 8 + c];
    rowinv[rrow] = 1.0f / s2;
  }
  __syncthreads();

  // ---- normalize in LDS + coalesced series store ----
  float* seriesBase = outSeries + ((size_t)bh * L_ + q0) * L_;
  for (int idx = tid; idx < 16 * 1024; idx += 128) {
    const int row = idx >> 10;
    const int col = idx & 1023;
    const float p = sm[row * LDSTRIDE + col] * rowinv[row];
    sm[row * LDSTRIDE + col] = p;
    seriesBase[(size_t)row * L_ + col] = p;
  }
  __syncthreads();

  // ---- V = P (16x1024, LDS) x values (1024x16 slice): one E-tile per wave ----
  {
    const int e0 = wave * 16;
    const float* vcol = V + ((size_t)b * L_ * H_ + h) * E_ + e0 + m;
    v8f acc = {};
    #pragma unroll 8
    for (int ki = 0; ki < 256; ++ki) {
      const int sb = ki * 4 + hi * 2;
      v2f bfrag;
      bfrag.x = vcol[(size_t)(sb + 0) * (H_ * E_)];
      bfrag.y = vcol[(size_t)(sb + 1) * (H_ * E_)];
      const v2f afrag = *(const v2f*)&sm[m * LDSTRIDE + sb];
      acc = __builtin_amdgcn_wmma_f32_16x16x4_f32(false, afrag, false, bfrag,
                                                  (short)0, acc, false, false);
    }
    #pragma unroll
    for (int r = 0; r < 8; ++r) {
      const int row = q0 + r + hi * 8;
      outV[(((size_t)b * L_ + row) * H_ + h) * E_ + e0 + m] = acc[r];
    }
  }
}

// ---------------------------------------------------------------------------
// Kernel 2: prior + sigma4 (pure bandwidth, 268 MB of writes).
// grid = B*H*L blocks of 256 threads; one block = one (b,h,i) row, 4 cols/thread.
// ---------------------------------------------------------------------------
__global__ __launch_bounds__(256)
void prior_sigma_kernel(const float* __restrict__ sigma,
                        float* __restrict__ prior,
                        float* __restrict__ sigma4) {
  const int i  = blockIdx.x & (L_ - 1);
  const int bh = blockIdx.x >> 10;
  const int b  = bh >> 3;
  const int h  = bh & 7;

  const float x = sigma[((size_t)b * L_ + i) * H_ + h];
  float s = 1.0f / (1.0f + __expf(-5.0f * x)) + 1e-5f;       // sigmoid(5x)+1e-5
  s = __expf(s * 1.0986122886681098f) - 1.0f;                // 3^s - 1
  const float inv2s2 = -1.0f / (2.0f * s * s);
  const float coef   = 0.3989422804014327f / s;              // 1/(sqrt(2pi)*s)

  const size_t base = ((size_t)bh * L_ + i) * L_;
  const int j0 = threadIdx.x * 4;
  v4f pr, sg;
  #pragma unroll
  for (int c = 0; c < 4; ++c) {
    const float d = (float)(i - (j0 + c));
    pr[c] = coef * __expf(d * d * inv2s2);
    sg[c] = s;
  }
  *(v4f*)(prior  + base + j0) = pr;
  *(v4f*)(sigma4 + base + j0) = sg;
}

// ---------------------------------------------------------------------------
extern "C" void kernel_launch(void* const* d_in, const int* in_sizes, int n_in,
                              void* d_out, int out_size, void* d_ws, size_t ws_size,
                              hipStream_t stream) {
  const float* Q     = (const float*)d_in[0];
  const float* K     = (const float*)d_in[1];
  const float* V     = (const float*)d_in[2];
  const float* sigma = (const float*)d_in[3];

  float* out = (float*)d_out;
  const size_t NV = (size_t)B_ * L_ * H_ * E_;       // 2,097,152
  const size_t NS = (size_t)B_ * H_ * L_ * L_;       // 33,554,432
  float* outV      = out;
  float* outSeries = out + NV;
  float* outPrior  = out + NV + NS;
  float* outSigma4 = out + NV + 2 * NS;

  const size_t shmem = (size_t)(16 * LDSTRIDE + 128 + 32) * sizeof(float); // ~66.4 KB

  attn_fused_kernel<<<B_ * H_ * (L_ / 16), 128, shmem, stream>>>(Q, K, V, outV, outSeries);
  prior_sigma_kernel<<<B_ * H_ * L_, 256, 0, stream>>>(sigma, outPrior, outSigma4);
}